// Attn_w_38757784879239
// MI455X (gfx1250) — compile-verified
//
#include <hip/hip_runtime.h>

// Sliding-window attention smoothing, fused single pass.
// rows=200000, RANK=64, WINDOW=11.

#define RANK   64
#define WIN    11
#define HALO   5                 // (WIN-1)/2
#define TILE   64                // rows per block
#define XROWS  (TILE + 2*HALO)   // 74 rows incl. halo
#define LD_X   66                // padded LDS row stride (floats) for x / var1 tiles
#define LD_W   68                // padded LDS row stride (floats) for W (float4-aligned)

typedef float v2f __attribute__((ext_vector_type(2)));
typedef float v8f __attribute__((ext_vector_type(8)));

// Hardware tanh (V_TANH_F32 is a TRANS32 op on gfx1250); guarded fallback.
static __device__ __forceinline__ float fast_tanh(float x) {
#if __has_builtin(__builtin_amdgcn_tanhf)
    return __builtin_amdgcn_tanhf(x);
#elif __has_builtin(__builtin_amdgcn_tanh_f32)
    return __builtin_amdgcn_tanh_f32(x);
#else
    return tanhf(x);
#endif
}

// exp(x) for x <= 0 via V_EXP_F32 (2^x): no range reduction needed.
static __device__ __forceinline__ float fast_exp_nonpos(float x) {
#if __has_builtin(__builtin_amdgcn_exp2f)
    return __builtin_amdgcn_exp2f(x * 1.44269504088896341f);
#else
    return __expf(x);
#endif
}

__global__ __launch_bounds__(128)
void attn_window_fused_kernel(const float* __restrict__ x,
                              const float* __restrict__ W,
                              const float* __restrict__ b,
                              float* __restrict__ out,
                              int nrows)
{
    __shared__ float xs[XROWS * LD_X];   // x tile + halo        (19536 B)
    __shared__ float vs[TILE  * LD_X];   // var1 = tanh(xW^T+b)  (16896 B)
    __shared__ float ws[RANK  * LD_W];   // W, row-major         (17408 B)
    __shared__ float bs[RANK];           // bias                 (  256 B)

    const int tid = threadIdx.x;
    const int r0  = blockIdx.x * TILE;

    // ---- Stage W (row-major: ws[n*LD_W+k] = W[n,k]) and bias into LDS ----
    for (int idx = tid; idx < RANK * (RANK / 4); idx += blockDim.x) {
        const int row = idx >> 4;        // 16 float4 per row
        const int c4  = idx & 15;
        float4 v = ((const float4*)W)[row * (RANK / 4) + c4];
        *(float4*)&ws[row * LD_W + c4 * 4] = v;   // LD_W*4 is 16B-aligned
    }
    if (tid < RANK) bs[tid] = b[tid];

    // ---- Stage x tile (with zero-padded halo) into LDS ----
    for (int idx = tid; idx < XROWS * (RANK / 4); idx += blockDim.x) {
        const int row = idx >> 4;
        const int c4  = idx & 15;
        const int g   = r0 - HALO + row;
        float4 v = make_float4(0.f, 0.f, 0.f, 0.f);
        if (g >= 0 && g < nrows) v = ((const float4*)x)[g * (RANK / 4) + c4];
        float* p = &xs[row * LD_X + c4 * 4];
        *(float2*)(p)     = make_float2(v.x, v.y);   // stride 66 -> 8B aligned
        *(float2*)(p + 2) = make_float2(v.z, v.w);
    }
    __syncthreads();

    // ---- Phase 2: var1 = tanh(x @ W^T + b) via V_WMMA_F32_16X16X4_F32 ----
    // Wave w owns rows [16w, 16w+16). A-frag layout (32-bit A 16x4):
    //   lanes 0-15: (M=lane, K=kb*4+{0,1}); lanes 16-31: (M=lane-16, K=kb*4+{2,3})
    // B = W^T so B[k][n] = W[n][k]: contiguous float2 from W's row n.
    {
        const int wave = tid >> 5;
        const int lane = tid & 31;
        const int m    = lane & 15;
        const int hi   = lane >> 4;               // 0 or 1
        const int arow = wave * 16 + m + HALO;    // tile row incl. halo offset

        v2f afrag[RANK / 4];
        #pragma unroll
        for (int kb = 0; kb < RANK / 4; ++kb)
            afrag[kb] = *(const v2f*)&xs[arow * LD_X + kb * 4 + hi * 2];

        #pragma unroll
        for (int nt = 0; nt < 4; ++nt) {
            v8f acc = {};
            const int n = nt * 16 + m;            // output column for this lane
            #pragma unroll
            for (int kb = 0; kb < RANK / 4; ++kb) {
                v2f bfrag = *(const v2f*)&ws[n * LD_W + kb * 4 + hi * 2];
                acc = __builtin_amdgcn_wmma_f32_16x16x4_f32(
                        false, afrag[kb], false, bfrag,
                        (short)0, acc, false, false);
            }
            const float bias = bs[n];
            // C/D layout: VGPR v -> (M = v + 8*hi, N = lane&15 (+16*nt))
            #pragma unroll
            for (int v = 0; v < 8; ++v) {
                const int row = wave * 16 + v + hi * 8;
                vs[row * LD_X + n] = fast_tanh(acc[v] + bias);
            }
        }
    }
    __syncthreads();

    // ---- Phase 3: window scores -> softmax -> weighted sum, 2 lanes/row ----
    // Written in 2-wide ext-vector form to lower to V_PK_FMA_F32.
    {
        const int r    = tid >> 1;        // block-local row 0..63
        const int half = tid & 1;         // each lane handles 32 columns
        const int g    = r0 + r;
        if (g < nrows) {
            const int c0 = half * 32;
            float s[WIN];
            #pragma unroll
            for (int w = 0; w < WIN; ++w) {
                v2f a0 = {0.f, 0.f}, a1 = {0.f, 0.f};
                #pragma unroll
                for (int c = 0; c < 32; c += 4) {
                    v2f x0 = *(const v2f*)&xs[(r + w) * LD_X + c0 + c];
                    v2f x1 = *(const v2f*)&xs[(r + w) * LD_X + c0 + c + 2];
                    v2f v0 = *(const v2f*)&vs[r * LD_X + c0 + c];
                    v2f v1 = *(const v2f*)&vs[r * LD_X + c0 + c + 2];
                    a0 = __builtin_elementwise_fma(x0, v0, a0);
                    a1 = __builtin_elementwise_fma(x1, v1, a1);
                }
                v2f acc = a0 + a1;
                float sv = acc.x + acc.y;
                sv += __shfl_xor(sv, 1, 32);        // combine the two halves
                s[w] = sv * 0.125f;                 // 1/sqrt(RANK)
            }
            float mx = s[0];
            #pragma unroll
            for (int w = 1; w < WIN; ++w) mx = fmaxf(mx, s[w]);
            float sum = 0.f;
            #pragma unroll
            for (int w = 0; w < WIN; ++w) { s[w] = fast_exp_nonpos(s[w] - mx); sum += s[w]; }
            const float inv = 1.0f / sum;

            #pragma unroll
            for (int c = 0; c < 32; c += 2) {
                v2f o = {0.f, 0.f};
                #pragma unroll
                for (int w = 0; w < WIN; ++w) {
                    v2f xv = *(const v2f*)&xs[(r + w) * LD_X + c0 + c];
                    v2f sw = {s[w], s[w]};
                    o = __builtin_elementwise_fma(sw, xv, o);
                }
                o *= inv;
                *(v2f*)&out[g * RANK + c0 + c] = o;
            }
        }
    }
}

extern "C" void kernel_launch(void* const* d_in, const int* in_sizes, int n_in,
                              void* d_out, int out_size, void* d_ws, size_t ws_size,
                              hipStream_t stream) {
    const float* x = (const float*)d_in[0];   // (rows, 64) f32
    const float* W = (const float*)d_in[1];   // (64, 64)   f32
    const float* b = (const float*)d_in[2];   // (64,)      f32
    float* out = (float*)d_out;               // (rows, 64) f32

    const int nrows = in_sizes[0] / RANK;
    const int grid  = (nrows + TILE - 1) / TILE;
    attn_window_fused_kernel<<<grid, 128, 0, stream>>>(x, W, b, out, nrows);
}